// GCN_62483184222721
// MI455X (gfx1250) — compile-verified
//
#include <hip/hip_runtime.h>

typedef float v2f __attribute__((ext_vector_type(2)));
typedef float v8f __attribute__((ext_vector_type(8)));

// ---------------------------------------------------------------- utilities
__global__ void k_fill(float* __restrict__ p, float v, int n) {
    int i = blockIdx.x * blockDim.x + threadIdx.x;
    if (i < n) p[i] = v;
}

__global__ void k_zero(float* __restrict__ p, int n) {
    for (int i = blockIdx.x * blockDim.x + threadIdx.x; i < n;
         i += gridDim.x * blockDim.x)
        p[i] = 0.0f;
}

// degree accumulation: deg pre-filled with 1.0 (self-loop), add 1 per in-edge
__global__ void k_deg_scatter(const int* __restrict__ dst, float* __restrict__ deg, int E) {
    int e = blockIdx.x * blockDim.x + threadIdx.x;
    if (e < E) unsafeAtomicAdd(&deg[dst[e]], 1.0f);
}

__global__ void k_dinv(const float* __restrict__ deg, float* __restrict__ dinv, int n) {
    int i = blockIdx.x * blockDim.x + threadIdx.x;
    if (i < n) {
        float d = deg[i];
        dinv[i] = (d > 0.0f) ? rsqrtf(d) : 0.0f;
    }
}

// ---------------------------------------------------------------- WMMA GEMM
// out[row, col] = dinv[row] * sum_k A[row,k] * W[k,col]
// One wave (32 lanes) computes one 16x16 tile via v_wmma_f32_16x16x4_f32.
// A frag (16x4 f32): lane L holds A[L&15][kbase + 2*(L>>4) + {0,1}]
// B frag (4x16 f32): lane L holds W[kbase + 2*(L>>4) + {0,1}][L&15]
// C/D (16x16 f32):   vgpr r, lane L -> row = r + 8*(L>>4), col = L&15
template <int K, int NF>
__launch_bounds__(128)
__global__ void k_gemm_scaled(const float* __restrict__ A,
                              const float* __restrict__ W,
                              const float* __restrict__ dinv,
                              float* __restrict__ out, int M) {
    const int lane  = threadIdx.x & 31;
    const int tileM = blockIdx.x;
    const int tileN = blockIdx.y * blockDim.y + threadIdx.y;
    const int r16   = lane & 15;
    const int hi    = lane >> 4;   // 0 or 1
    const int kh    = hi * 2;      // k sub-offset within the 4-wide slab

    int rowA = tileM * 16 + r16;
    if (rowA >= M) rowA = M - 1;             // clamp loads; EXEC stays all-ones
    const int colB = tileN * 16 + r16;

    const float* ap = A + (size_t)rowA * K + kh;
    const float* bp = W + (size_t)kh * NF + colB;

    __builtin_prefetch(bp, 0, 3);            // global_prefetch_b8 on W panel

    v8f c = {};
#pragma unroll 8
    for (int k = 0; k < K; k += 4) {
        v2f a = *(const v2f*)ap;             // contiguous pair A[row][k+kh..+1]
        v2f b;
        b.x = bp[0];                         // W[k+kh  ][col]
        b.y = bp[NF];                        // W[k+kh+1][col]
        c = __builtin_amdgcn_wmma_f32_16x16x4_f32(
                /*neg_a=*/false, a, /*neg_b=*/false, b,
                /*c_mod=*/(short)0, c, /*reuse_a=*/false, /*reuse_b=*/false);
        ap += 4;
        bp += 4 * NF;
    }

    // epilogue: scale rows by dinv and store; full-tile fast path
    const int rowBase = tileM * 16 + hi * 8;
    float* op = out + (size_t)rowBase * NF + colB;
    if (rowBase + 8 <= M) {
        const float4 d0 = *(const float4*)(dinv + rowBase);
        const float4 d1 = *(const float4*)(dinv + rowBase + 4);
        op[0 * (size_t)NF] = c[0] * d0.x;
        op[1 * (size_t)NF] = c[1] * d0.y;
        op[2 * (size_t)NF] = c[2] * d0.z;
        op[3 * (size_t)NF] = c[3] * d0.w;
        op[4 * (size_t)NF] = c[4] * d1.x;
        op[5 * (size_t)NF] = c[5] * d1.y;
        op[6 * (size_t)NF] = c[6] * d1.z;
        op[7 * (size_t)NF] = c[7] * d1.w;
    } else {
#pragma unroll
        for (int r = 0; r < 8; ++r) {
            int row = rowBase + r;
            if (row < M) op[(size_t)r * NF] = c[r] * dinv[row];
        }
    }
}

// ---------------------------------------------------------------- scatter-add
// agg[dst[e]][f] += hs[src[e]][f]; one thread per feature-quad:
// one global_load_b128 + four global_atomic_add_f32 per thread.
template <int F>
__global__ void k_edge_scatter(const float* __restrict__ hs,
                               const int* __restrict__ src,
                               const int* __restrict__ dst,
                               float* __restrict__ agg, int E) {
    constexpr int TPE = F / 4;                           // threads per edge
    int e = blockIdx.x * (256 / TPE) + threadIdx.x / TPE;
    if (e >= E) return;
    int q = (threadIdx.x % TPE) * 4;                     // feature quad
    const float4 v = *(const float4*)(hs + (size_t)src[e] * F + q);
    float* a = agg + (size_t)dst[e] * F + q;
    unsafeAtomicAdd(a + 0, v.x);
    unsafeAtomicAdd(a + 1, v.y);
    unsafeAtomicAdd(a + 2, v.z);
    unsafeAtomicAdd(a + 3, v.w);
}

// ---------------------------------------------------------------- finalize
// out = [relu]( dinv[i] * (agg + hs) + bias[f] ), vectorized by quads.
template <int F, bool RELU>
__global__ void k_finalize(const float* __restrict__ agg,
                           const float* __restrict__ hs,
                           const float* __restrict__ dinv,
                           const float* __restrict__ bias,
                           float* __restrict__ out, int N) {
    constexpr int TPR = F / 4;                           // quads per row
    int t = blockIdx.x * blockDim.x + threadIdx.x;
    if (t >= N * TPR) return;
    int i = t / TPR;
    int q = (t % TPR) * 4;
    size_t off = (size_t)i * F + q;
    float di = dinv[i];
    const float4 va = *(const float4*)(agg + off);
    const float4 vh = *(const float4*)(hs + off);
    const float4 vb = *(const float4*)(bias + q);
    float4 r;
    r.x = di * (va.x + vh.x) + vb.x;
    r.y = di * (va.y + vh.y) + vb.y;
    r.z = di * (va.z + vh.z) + vb.z;
    r.w = di * (va.w + vh.w) + vb.w;
    if (RELU) {
        r.x = fmaxf(r.x, 0.0f);
        r.y = fmaxf(r.y, 0.0f);
        r.z = fmaxf(r.z, 0.0f);
        r.w = fmaxf(r.w, 0.0f);
    }
    *(float4*)(out + off) = r;
}

// ---------------------------------------------------------------- launch
extern "C" void kernel_launch(void* const* d_in, const int* in_sizes, int n_in,
                              void* d_out, int out_size, void* d_ws, size_t ws_size,
                              hipStream_t stream) {
    const float* x  = (const float*)d_in[0];
    const float* W1 = (const float*)d_in[1];
    const float* b1 = (const float*)d_in[2];
    const float* W2 = (const float*)d_in[3];
    const float* b2 = (const float*)d_in[4];
    const int*   ei = (const int*)d_in[5];

    const int IN = 128, HID = 128, OUT = 64;
    const int N = in_sizes[0] / IN;
    const int E = in_sizes[5] / 2;
    const int* src = ei;        // edge_index[0]
    const int* dst = ei + E;    // edge_index[1]

    float* ws   = (float*)d_ws;
    float* deg  = ws;                           // N
    float* dinv = deg + N;                      // N
    float* bufA = dinv + N;                     // N*HID  (scaled activations hs)
    float* bufB = bufA + (size_t)N * HID;       // N*HID  (edge aggregation)
    float* bufC = bufB + (size_t)N * HID;       // N*HID  (layer-1 relu output)
    float* out  = (float*)d_out;                // N*OUT

    // --- symmetric normalization coefficients
    k_fill<<<(N + 255) / 256, 256, 0, stream>>>(deg, 1.0f, N);   // self-loop
    k_deg_scatter<<<(E + 255) / 256, 256, 0, stream>>>(dst, deg, E);
    k_dinv<<<(N + 255) / 256, 256, 0, stream>>>(deg, dinv, N);

    const int tilesM = (N + 15) / 16;   // 625 for N=10000 (exact)

    // --- layer 1: hs1 = dinv .* (x @ W1); agg over edges; relu finalize
    k_zero<<<1024, 256, 0, stream>>>(bufB, N * HID);
    {
        dim3 blk(32, 4);
        dim3 grd(tilesM, HID / 64);     // 2 block-columns of 4 waves
        k_gemm_scaled<128, 128><<<grd, blk, 0, stream>>>(x, W1, dinv, bufA, N);
    }
    k_edge_scatter<128><<<(E + 7) / 8, 256, 0, stream>>>(bufA, src, dst, bufB, E);
    k_finalize<128, true><<<(N * 32 + 255) / 256, 256, 0, stream>>>(
        bufB, bufA, dinv, b1, bufC, N);

    // --- layer 2: hs2 = dinv .* (h1 @ W2); agg over edges; finalize -> d_out
    k_zero<<<1024, 256, 0, stream>>>(bufB, N * OUT);
    {
        dim3 blk(32, 4);
        dim3 grd(tilesM, OUT / 64);     // 1 block-column
        k_gemm_scaled<128, 64><<<grd, blk, 0, stream>>>(bufC, W2, dinv, bufA, N);
    }
    k_edge_scatter<64><<<(E + 15) / 16, 256, 0, stream>>>(bufA, src, dst, bufB, E);
    k_finalize<64, false><<<(N * 16 + 255) / 256, 256, 0, stream>>>(
        bufB, bufA, dinv, b2, out, N);
}